// NSLoss_47175920779678
// MI455X (gfx1250) — compile-verified
//
#include <hip/hip_runtime.h>

// NSLoss (negative-sampling loss) for MI455X / gfx1250, wave32.
//
// Roofline: ~0.29 GFLOP vs ~88 MB of HBM traffic (embs 32MB streamed once,
// 51MB weight table resident in 192MB L2, indices ~4.5MB) -> memory bound at
// ~4us. Keep everything f32 (matches reference numerics; conversion cannot
// reduce HBM bytes) and use V_WMMA_F32_16X16X4_F32 for the 16 negative dot
// products per sample: A = embedding chunk broadcast to all 16 rows
// (documented A layout), B = 16 gathered negative weight rows as columns.
// All rows of D are identical, so each lane reads its neg score from acc[0].

typedef __attribute__((ext_vector_type(2))) float v2f;
typedef __attribute__((ext_vector_type(4))) float v4f;
typedef __attribute__((ext_vector_type(8))) float v8f;

#define EMB               128
#define NEG               16
#define WAVES_PER_BLOCK   8
#define SAMPLES_PER_WAVE  4
#define BLOCK_THREADS     256   // 8 wave32 waves

__device__ __forceinline__ float log_sigmoid(float x) {
    // log(sigmoid(x)) = min(x,0) - log1p(exp(-|x|))  (numerically stable)
    return fminf(x, 0.0f) - log1pf(__expf(-fabsf(x)));
}

__global__ void nsloss_main(const float* __restrict__ embs,
                            const int*   __restrict__ label,
                            const int*   __restrict__ negs,
                            const float* __restrict__ weights,
                            float*       __restrict__ blockPartials,
                            int nTotal) {
    __shared__ float ldsE[WAVES_PER_BLOCK][EMB];   // per-wave embedding row
    __shared__ float wavePartial[WAVES_PER_BLOCK];

    const int tid  = threadIdx.x;
    const int wave = tid >> 5;          // wave32
    const int lane = tid & 31;
    const int halfSel = (lane < 16) ? 0 : 2;  // K sub-block owned by this lane half
    const int col  = lane & 15;               // D column / neg slot owned by lane

    float waveLoss = 0.0f;
    const int base = (blockIdx.x * WAVES_PER_BLOCK + wave) * SAMPLES_PER_WAVE;

    for (int s = 0; s < SAMPLES_PER_WAVE; ++s) {
        const int i = base + s;               // wave-uniform
        if (i >= nTotal) break;               // wave-uniform branch: EXEC stays full

        // ---- stage embedding row: lane owns dims [4*lane, 4*lane+3] ----
        v4f e4 = *(const v4f*)(embs + (size_t)i * EMB + lane * 4);
        *(v4f*)(&ldsE[wave][lane * 4]) = e4;  // same-wave LDS, DScnt-ordered

        // ---- positive score: plain f32 dot + wave32 butterfly reduce ----
        const int lbl = label[i];
        v4f w4 = *(const v4f*)(weights + (size_t)lbl * EMB + lane * 4);
        float p = e4.x * w4.x + e4.y * w4.y + e4.z * w4.z + e4.w * w4.w;
        #pragma unroll
        for (int off = 16; off >= 1; off >>= 1)
            p += __shfl_xor(p, off, 32);

        // ---- negative scores via V_WMMA_F32_16X16X4_F32, K=128 in 32 steps ----
        // B column n = weights[negs[i][n]] chunk; lane supplies column (lane&15),
        // K rows {halfSel, halfSel+1} per the lane-half split (mirrors A layout).
        const int nrow = negs[i * NEG + col];
        const float* wrow = weights + (size_t)nrow * EMB;

        v8f acc = {};
        #pragma unroll
        for (int t = 0; t < EMB / 4; ++t) {
            const int c = t * 4 + halfSel;
            // A: 16x4, all 16 rows = embedding chunk (broadcast via LDS).
            // Documented layout: lane half 0 holds K={0,1}, half 1 holds K={2,3}.
            v2f a = *(const v2f*)(&ldsE[wave][c]);
            // B: 4x16, column (lane&15) = this lane's neg weight row chunk.
            v2f b = *(const v2f*)(wrow + c);
            acc = __builtin_amdgcn_wmma_f32_16x16x4_f32(
                /*neg_a=*/false, a, /*neg_b=*/false, b,
                /*c_mod=*/(short)0, acc, /*reuse_a=*/false, /*reuse_b=*/false);
        }

        // All D rows identical -> lane L's acc[0] = dot(e, W[negs[i][L&15]]).
        // neg_score = -dot; reduce log_sigmoid over lanes (each score counted
        // twice, once per lane half -> exact *0.5 fixup).
        float ls = log_sigmoid(-acc[0]);
        #pragma unroll
        for (int off = 16; off >= 1; off >>= 1)
            ls += __shfl_xor(ls, off, 32);
        const float sum_log_sampled = 0.5f * ls;

        waveLoss += log_sigmoid(p) + sum_log_sampled;
    }

    if (lane == 0) wavePartial[wave] = waveLoss;
    __syncthreads();
    if (tid == 0) {
        float t = 0.0f;
        #pragma unroll
        for (int w = 0; w < WAVES_PER_BLOCK; ++w) t += wavePartial[w];
        blockPartials[blockIdx.x] = t;
    }
}

// Deterministic fixed-order final reduction (no float atomics).
__global__ void nsloss_reduce(const float* __restrict__ partials, int nPartials,
                              float* __restrict__ out, float invN) {
    __shared__ float sm[256];
    const int tid = threadIdx.x;
    float s = 0.0f;
    for (int idx = tid; idx < nPartials; idx += 256)
        s += partials[idx];
    sm[tid] = s;
    __syncthreads();
    #pragma unroll
    for (int off = 128; off >= 1; off >>= 1) {
        if (tid < off) sm[tid] += sm[tid + off];
        __syncthreads();
    }
    if (tid == 0) out[0] = -sm[0] * invN;
}

extern "C" void kernel_launch(void* const* d_in, const int* in_sizes, int n_in,
                              void* d_out, int out_size, void* d_ws, size_t ws_size,
                              hipStream_t stream) {
    // setup_inputs order: input(f32)[N], embs(f32)[N,128], label(i32)[N],
    // negs(i32)[N,16], weights(f32)[NUM_NODES,128]
    const float* embs    = (const float*)d_in[1];
    const int*   label   = (const int*)  d_in[2];
    const int*   negs    = (const int*)  d_in[3];
    const float* weights = (const float*)d_in[4];
    const int n = in_sizes[0];

    const int samplesPerBlock = WAVES_PER_BLOCK * SAMPLES_PER_WAVE;  // 32
    const int nBlocks = (n + samplesPerBlock - 1) / samplesPerBlock; // 2048 for N=65536

    float* partials = (float*)d_ws;  // nBlocks floats (8KB) << ws_size

    nsloss_main<<<nBlocks, BLOCK_THREADS, 0, stream>>>(
        embs, label, negs, weights, partials, n);
    nsloss_reduce<<<1, 256, 0, stream>>>(
        partials, nBlocks, (float*)d_out, 1.0f / (float)n);
}